// GraphConvolution_1726576853271
// MI455X (gfx1250) — compile-verified
//
#include <hip/hip_runtime.h>

// ---------------------------------------------------------------------------
// MI455X (gfx1250) RGCN-style graph convolution
//   Out[4096,32] = A[4096,65536] @ ( X[4096,64] @ (comp[16,8] . W_F[8,64,32]) )
// Memory-bound on the single 1.07 GB streaming read of A (floor ~46us at
// 23.3 TB/s).  All matmuls use V_WMMA_F32_16X16X4_F32 (wave32 WMMA).
// Kernel 3 stages tiles with GLOBAL_LOAD_ASYNC_TO_LDS_B128 (ASYNCcnt) and
// double-buffers them.
// ---------------------------------------------------------------------------

typedef float v2f __attribute__((ext_vector_type(2)));
typedef float v4f __attribute__((ext_vector_type(4)));
typedef float v8f __attribute__((ext_vector_type(8)));

#define NN   4096
#define RR   16
#define BB   8
#define DIN  64
#define DOUT 32
#define RN   (RR * NN)          // 65536 = K of the big GEMM

#if defined(__has_builtin)
#if __has_builtin(__builtin_amdgcn_global_load_async_to_lds_b128) && \
    __has_builtin(__builtin_amdgcn_s_wait_asynccnt)
#define USE_ASYNC 1
#endif
#endif

// ---------------------------------------------------------------------------
// Kernel 1: W[r][j][k] = sum_b comp[r][b] * W_F[b][j][k]   (16x64x32 output)
// ---------------------------------------------------------------------------
__global__ __launch_bounds__(256) void k_build_w(const float* __restrict__ wf,
                                                 const float* __restrict__ comp,
                                                 float* __restrict__ W) {
    int idx = blockIdx.x * 256 + threadIdx.x;   // 0..32767
    int r   = idx >> 11;
    int jk  = idx & 2047;
    float acc = 0.0f;
#pragma unroll
    for (int b = 0; b < BB; ++b)
        acc += comp[r * BB + b] * wf[b * (DIN * DOUT) + jk];
    W[idx] = acc;
}

// ---------------------------------------------------------------------------
// Kernel 2: FW[r*N + n][k] = X[n][:] @ W[r][:][k]      (65536 x 32, K = 64)
// One wave per 16-row tile; two 16x16 WMMA accumulators.  Traffic is tiny
// (X read 16x = 16 MB, all L2-resident), so direct global loads.
//
// WMMA f32 16x16x4 fragment layout (ISA 7.12.2):
//   A (16x4):  lane = M (mod 16), K = vgpr + 2*(lane>=16)
//   B (4x16):  lane = N (mod 16), K = vgpr + 2*(lane>=16)   [mirrored]
//   C/D 16x16: lane = N (mod 16), M = vgpr + 8*(lane>=16)
// ---------------------------------------------------------------------------
__global__ __launch_bounds__(128) void k_fw(const float* __restrict__ X,
                                            const float* __restrict__ W,
                                            float* __restrict__ FW) {
    int wave = blockIdx.x * 4 + (threadIdx.x >> 5);   // 0..4095 row tiles
    int l = threadIdx.x & 31;
    int h = l >> 4;
    int m = l & 15;
    int r    = wave >> 8;
    int nrow = ((wave & 255) << 4) + m;
    const float* wr = W + r * (DIN * DOUT);

    v8f c0 = {}; v8f c1 = {};
#pragma unroll
    for (int kk = 0; kk < DIN; kk += 4) {
        int ka = kk + 2 * h;
        v2f a = *(const v2f*)(X + nrow * DIN + ka);
        v2f b0, b1;
        b0.x = wr[ka * DOUT + m];
        b0.y = wr[(ka + 1) * DOUT + m];
        b1.x = wr[ka * DOUT + 16 + m];
        b1.y = wr[(ka + 1) * DOUT + 16 + m];
        c0 = __builtin_amdgcn_wmma_f32_16x16x4_f32(false, a, false, b0, (short)0, c0, false, false);
        c1 = __builtin_amdgcn_wmma_f32_16x16x4_f32(false, a, false, b1, (short)0, c1, false, false);
    }

    float* out = FW + (size_t)(wave * 16) * DOUT;
#pragma unroll
    for (int v = 0; v < 8; ++v) {
        int M = v + 8 * h;
        out[M * DOUT + m]      = c0[v];
        out[M * DOUT + 16 + m] = c1[v];
    }
}

// ---------------------------------------------------------------------------
// Kernel 3: Out[4096,32] = A[4096,65536] @ FW[65536,32]
// One block (8 waves) per 16-row output tile; wave w owns K segment
// [w*8192,(w+1)*8192).  Deterministic split-K reduction through LDS.
// ---------------------------------------------------------------------------
#define SEG  8
#define SEGK (RN / SEG)          // 8192 K per wave

#ifdef USE_ASYNC
// ======================= async + double-buffered path =======================
// K tile = 16.  Per wave: 2 buffers x (A 16x20 + FW 16x40) floats = 7680 B;
// 8 waves = 60 KB static LDS (under the 64 KB static cap).
// Bank tuning (64 banks x 4B):
//   lA stride 20 = 4*5: A-frag b64 reads -> lanes 0-15 banks {4q,4q+1},
//     lanes 16-31 {4q+2,4q+3}: conflict-free.
//   lB stride 40: 2*40 mod 64 = 16 -> the two half-wave b32 reads use
//     disjoint 16-bank ranges: conflict-free.

// Builtin signature (from the round-2 diagnostic): pointee is a 4 x i32
// gcc-vector; global side is AS1 (printed '__device__'), LDS side AS3.
typedef int v4i __attribute__((vector_size(16)));
typedef __attribute__((address_space(1))) v4i g_v4i;
typedef __attribute__((address_space(3))) v4i l_v4i;

__device__ __forceinline__ static g_v4i* as_glob(const void* p) {
    return (g_v4i*)(unsigned long long)p;
}
__device__ __forceinline__ static l_v4i* as_lds(void* p) {
    // generic LDS pointer: addr[31:0] is the LDS byte offset (ISA 10.2)
    return (l_v4i*)(unsigned int)(unsigned long long)p;
}

#define KT    16
#define LA_S  20
#define LB_S  40
#define LA_F  (16 * LA_S)        // 320 floats
#define LB_F  (KT * LB_S)        // 640 floats
#define BUF_F (LA_F + LB_F)      // 960 floats per buffer
#define WAVE_F (2 * BUF_F)       // 1920 floats per wave

__global__ __launch_bounds__(256) void k_agg(const float* __restrict__ A,
                                             const float* __restrict__ FW,
                                             float* __restrict__ out) {
    __shared__ float smem[SEG * WAVE_F];   // 61440 B; reused for reduction

    const int tid = threadIdx.x;
    const int w   = tid >> 5;
    const int l   = tid & 31;
    const int h   = l >> 4;
    const int m   = l & 15;
    const int rt  = blockIdx.x;            // 0..255 row tile

    float* wbase = smem + w * WAVE_F;
    const int kseg = w * SEGK;
    const int NT   = SEGK / KT;            // 512 tiles per wave

    // Issue one tile's 6 async b128 loads (2 for A, 4 for FW), per lane 16 B.
    auto issue = [&](int t, int buf) {
        const int kbase = kseg + t * KT;
        float* bA = wbase + buf * BUF_F;
        float* bB = bA + LA_F;
#pragma unroll
        for (int i = 0; i < 2; ++i) {          // A tile: 16 rows x 16 K
            int f = i * 32 + l;                // 16B chunk id, 4 chunks/row
            int row = f >> 2, col4 = (f & 3) << 2;
            __builtin_amdgcn_global_load_async_to_lds_b128(
                as_glob(A + (size_t)(rt * 16 + row) * RN + kbase + col4),
                as_lds(bA + row * LA_S + col4), 0, 0);
        }
#pragma unroll
        for (int i = 0; i < 4; ++i) {          // FW tile: 16 K-rows x 32 cols
            int f = i * 32 + l;                // 8 chunks/row
            int krow = f >> 3, col4 = (f & 7) << 2;
            __builtin_amdgcn_global_load_async_to_lds_b128(
                as_glob(FW + (size_t)(kbase + krow) * DOUT + col4),
                as_lds(bB + krow * LB_S + col4), 0, 0);
        }
    };

    v8f c0 = {}; v8f c1 = {};

    issue(0, 0);                               // prologue prefetch
    for (int t = 0; t < NT; ++t) {
        if (t + 1 < NT) {
            // WAR fence: prior tile's ds_loads (DScnt domain) must land
            // before the async engine overwrites that buffer.
            asm volatile("s_wait_dscnt 0x0" ::: "memory");
            issue(t + 1, (t + 1) & 1);
            // async loads complete in order: <=6 outstanding means tile t
            // (issued earlier) has fully landed in LDS.
            __builtin_amdgcn_s_wait_asynccnt(6);
        } else {
            __builtin_amdgcn_s_wait_asynccnt(0);
        }
        asm volatile("" ::: "memory");         // keep LDS reads below the wait

        float* bA = wbase + (t & 1) * BUF_F;
        float* bB = bA + LA_F;
#pragma unroll
        for (int kk = 0; kk < KT; kk += 4) {
            int ka = kk + 2 * h;
            v2f a = *(const v2f*)(bA + m * LA_S + ka);
            v2f b0, b1;
            b0.x = bB[ka * LB_S + m];
            b0.y = bB[(ka + 1) * LB_S + m];
            b1.x = bB[ka * LB_S + 16 + m];
            b1.y = bB[(ka + 1) * LB_S + 16 + m];
            c0 = __builtin_amdgcn_wmma_f32_16x16x4_f32(false, a, false, b0, (short)0, c0, false, false);
            c1 = __builtin_amdgcn_wmma_f32_16x16x4_f32(false, a, false, b1, (short)0, c1, false, false);
        }
    }

    __syncthreads();                           // all waves done with tiles
    float* red = smem;                         // red[8][16][32] = 4096 floats
#pragma unroll
    for (int v = 0; v < 8; ++v) {
        int M = v + 8 * h;
        red[w * 512 + M * 32 + m]      = c0[v];
        red[w * 512 + M * 32 + 16 + m] = c1[v];
    }
    __syncthreads();

    int e = tid * 2;
    float s0 = 0.0f, s1 = 0.0f;
#pragma unroll
    for (int ww = 0; ww < SEG; ++ww) {
        s0 += red[ww * 512 + e];
        s1 += red[ww * 512 + e + 1];
    }
    out[rt * 512 + e]     = s0;
    out[rt * 512 + e + 1] = s1;
}

#else
// ===================== fallback: synchronous staging ========================
#define KT        32
#define LA_STRIDE 36
#define LB_STRIDE 40
#define LA_FLOATS (16 * LA_STRIDE)
#define LB_FLOATS (KT * LB_STRIDE)
#define WAVE_SMEM (LA_FLOATS + LB_FLOATS)

__global__ __launch_bounds__(256) void k_agg(const float* __restrict__ A,
                                             const float* __restrict__ FW,
                                             float* __restrict__ out) {
    __shared__ float smem[SEG * WAVE_SMEM];

    const int tid = threadIdx.x;
    const int w   = tid >> 5;
    const int l   = tid & 31;
    const int h   = l >> 4;
    const int m   = l & 15;
    const int rt  = blockIdx.x;

    float* lA = smem + w * WAVE_SMEM;
    float* lB = lA + LA_FLOATS;

    v8f c0 = {}; v8f c1 = {};
    const int kseg = w * SEGK;

    for (int t = 0; t < SEGK / KT; ++t) {
        const int kbase = kseg + t * KT;
#pragma unroll
        for (int i = 0; i < 4; ++i) {
            int f = i * 32 + l;
            int row = f >> 3, col4 = (f & 7) << 2;
            v4f q = *(const v4f*)(A + (size_t)(rt * 16 + row) * RN + kbase + col4);
            *(v4f*)(lA + row * LA_STRIDE + col4) = q;
        }
#pragma unroll
        for (int i = 0; i < 8; ++i) {
            int f = i * 32 + l;
            int krow = f >> 3, col4 = (f & 7) << 2;
            v4f q = *(const v4f*)(FW + (size_t)(kbase + krow) * DOUT + col4);
            *(v4f*)(lB + krow * LB_STRIDE + col4) = q;
        }
        __syncthreads();
#pragma unroll
        for (int kk = 0; kk < KT; kk += 4) {
            int ka = kk + 2 * h;
            v2f a = *(const v2f*)(lA + m * LA_STRIDE + ka);
            v2f b0, b1;
            b0.x = lB[ka * LB_STRIDE + m];
            b0.y = lB[(ka + 1) * LB_STRIDE + m];
            b1.x = lB[ka * LB_STRIDE + 16 + m];
            b1.y = lB[(ka + 1) * LB_STRIDE + 16 + m];
            c0 = __builtin_amdgcn_wmma_f32_16x16x4_f32(false, a, false, b0, (short)0, c0, false, false);
            c1 = __builtin_amdgcn_wmma_f32_16x16x4_f32(false, a, false, b1, (short)0, c1, false, false);
        }
        __syncthreads();
    }

    float* red = smem;
#pragma unroll
    for (int v = 0; v < 8; ++v) {
        int M = v + 8 * h;
        red[w * 512 + M * 32 + m]      = c0[v];
        red[w * 512 + M * 32 + 16 + m] = c1[v];
    }
    __syncthreads();

    int e = tid * 2;
    float s0 = 0.0f, s1 = 0.0f;
#pragma unroll
    for (int ww = 0; ww < SEG; ++ww) {
        s0 += red[ww * 512 + e];
        s1 += red[ww * 512 + e + 1];
    }
    out[rt * 512 + e]     = s0;
    out[rt * 512 + e + 1] = s1;
}
#endif  // USE_ASYNC

// ---------------------------------------------------------------------------
// Host launcher.  Inputs (setup_inputs order): X, A, W_F, W_F_comp.
// Workspace: FW (8 MB) then W (128 KB).
// ---------------------------------------------------------------------------
extern "C" void kernel_launch(void* const* d_in, const int* in_sizes, int n_in,
                              void* d_out, int out_size, void* d_ws, size_t ws_size,
                              hipStream_t stream) {
    const float* X    = (const float*)d_in[0];
    const float* A    = (const float*)d_in[1];
    const float* WF   = (const float*)d_in[2];
    const float* COMP = (const float*)d_in[3];

    float* FW = (float*)d_ws;                         // 65536*32 floats
    float* W  = FW + (size_t)RN * DOUT;               // 16*64*32 floats

    k_build_w<<<128, 256, 0, stream>>>(WF, COMP, W);
    k_fw<<<1024, 128, 0, stream>>>(X, W, FW);
    k_agg<<<256, 256, 0, stream>>>(A, FW, (float*)d_out);
}